// MultiHeadAttentionPoolingLayer_66022237274247
// MI455X (gfx1250) — compile-verified
//
#include <hip/hip_runtime.h>

typedef float v2f __attribute__((ext_vector_type(2)));
typedef float v8f __attribute__((ext_vector_type(8)));

#define K1_WAVES 8          // waves per block in the score kernel
#define NEG_HUGE (-3.402823466e38f)

// -----------------------------------------------------------------------------
// Kernel 1: scores[n][h] = (sum_d x[n][d] * W[h][d] + b[h]) / T
// WMMA f32 16x16x4:  A = W rows (M=head; LDS copy zero-padded to 16 rows so the
// A fragment feeds the WMMA directly with no masking VALU op), B = x^T tile
// (K x N, N = node). Result D[m][n]: lanes 0-15 hold head m = accIdx (0..3)
// for node col n = lane, giving a single coalesced float4 store per node.
// -----------------------------------------------------------------------------
__global__ void scores_wmma_kernel(const float* __restrict__ x,
                                   const float* __restrict__ W,
                                   const float* __restrict__ bias,
                                   const float* __restrict__ temp,
                                   float* __restrict__ scores,
                                   int N, int D, int H) {
    __shared__ float Wlds[16 * 256];             // 16 rows x D, rows >= H zeroed
    const int tid = threadIdx.x;
    for (int i = tid; i < 16 * D; i += blockDim.x)
        Wlds[i] = (i < H * D) ? W[i] : 0.0f;
    __syncthreads();

    const int wave   = tid >> 5;
    const int lane   = tid & 31;
    const int lane15 = lane & 15;
    const bool hi    = lane >= 16;
    const int chunk  = blockIdx.x * K1_WAVES + wave;
    const int node_base = chunk * 16;
    if (node_base >= N) return;                  // wave-uniform exit

    const int  noderaw = node_base + lane15;
    const int  node    = (noderaw < N) ? noderaw : (N - 1);   // clamp loads
    const size_t xrow  = (size_t)node * (size_t)D;

    const int koff = hi ? 2 : 0;                 // K pair per half-wave
    const float* __restrict__ wrow = &Wlds[lane15 * D];

    v8f acc = {0.f, 0.f, 0.f, 0.f, 0.f, 0.f, 0.f, 0.f};
    const int ksteps = D >> 2;                   // K = 4 per WMMA
    for (int kk = 0; kk < ksteps; ++kk) {
        const int klo = (kk << 2) + koff;
        const float2 wv = *(const float2*)&wrow[klo];                   // A frag
        const float2 xv = *(const float2*)&x[xrow + (size_t)klo];       // B frag
        v2f a;  a[0]  = wv.x;  a[1]  = wv.y;
        v2f bb; bb[0] = xv.x;  bb[1] = xv.y;
        acc = __builtin_amdgcn_wmma_f32_16x16x4_f32(
                  /*neg_a=*/false, a, /*neg_b=*/false, bb,
                  /*c_mod=*/(short)0, acc, /*reuse_a=*/false, /*reuse_b=*/false);
    }

    // lanes 0-15: acc[h] = score(head h, node = node_base + lane)
    if (!hi && noderaw < N) {
        const float invT = 1.0f / temp[0];
        float4 o;
        o.x = (acc[0] + bias[0]) * invT;
        o.y = (acc[1] + bias[1]) * invT;
        o.z = (acc[2] + bias[2]) * invT;
        o.w = (acc[3] + bias[3]) * invT;
        *(float4*)&scores[(size_t)noderaw * 4] = o;
    }
}

// -----------------------------------------------------------------------------
__device__ __forceinline__ int lower_bound_i(const int* __restrict__ a, int n, int v) {
    int lo = 0, hi = n;
    while (lo < hi) { int mid = (lo + hi) >> 1; if (a[mid] < v) lo = mid + 1; else hi = mid; }
    return lo;
}

// Kernel 2: per-graph, per-head max and exp-sum (one wave32 per graph).
__global__ void seg_stats_kernel(const float* __restrict__ scores,
                                 const int* __restrict__ bi,
                                 float* __restrict__ gmax,
                                 float* __restrict__ gden,
                                 int N) {
    const int g    = blockIdx.x;
    const int lane = threadIdx.x;                 // 32 threads
    const int start = lower_bound_i(bi, N, g);
    const int end   = lower_bound_i(bi, N, g + 1);
    if (start >= end) {                           // empty graph: benign defaults
        if (lane == 0) {
            float4 m = {NEG_HUGE, NEG_HUGE, NEG_HUGE, NEG_HUGE};
            float4 z = {0.f, 0.f, 0.f, 0.f};
            *(float4*)&gmax[(size_t)g * 4] = m;
            *(float4*)&gden[(size_t)g * 4] = z;
        }
        return;
    }
    float m0 = NEG_HUGE, m1 = NEG_HUGE, m2 = NEG_HUGE, m3 = NEG_HUGE;
    for (int n = start + lane; n < end; n += 32) {
        float4 s = *(const float4*)&scores[(size_t)n * 4];
        m0 = fmaxf(m0, s.x); m1 = fmaxf(m1, s.y);
        m2 = fmaxf(m2, s.z); m3 = fmaxf(m3, s.w);
    }
    for (int off = 16; off; off >>= 1) {
        m0 = fmaxf(m0, __shfl_xor(m0, off));
        m1 = fmaxf(m1, __shfl_xor(m1, off));
        m2 = fmaxf(m2, __shfl_xor(m2, off));
        m3 = fmaxf(m3, __shfl_xor(m3, off));
    }
    float d0 = 0.f, d1 = 0.f, d2 = 0.f, d3 = 0.f;
    for (int n = start + lane; n < end; n += 32) {
        float4 s = *(const float4*)&scores[(size_t)n * 4];
        d0 += __expf(s.x - m0); d1 += __expf(s.y - m1);
        d2 += __expf(s.z - m2); d3 += __expf(s.w - m3);
    }
    for (int off = 16; off; off >>= 1) {
        d0 += __shfl_xor(d0, off); d1 += __shfl_xor(d1, off);
        d2 += __shfl_xor(d2, off); d3 += __shfl_xor(d3, off);
    }
    if (lane == 0) {
        float4 m = {m0, m1, m2, m3};
        float4 d = {d0, d1, d2, d3};
        *(float4*)&gmax[(size_t)g * 4] = m;
        *(float4*)&gden[(size_t)g * 4] = d;
    }
}

// Kernel 3: attn[n][h] = exp(score - max)/den ; attn_out is [H, N];
// wsum[n] = (1/H) * sum_h attn[n][h]
__global__ void attn_kernel(const float* __restrict__ scores,
                            const int* __restrict__ bi,
                            const float* __restrict__ gmax,
                            const float* __restrict__ gden,
                            float* __restrict__ attn_out,
                            float* __restrict__ wsum,
                            int N, float invH) {
    const int n = blockIdx.x * blockDim.x + threadIdx.x;
    if (n >= N) return;
    const int g = bi[n];
    float4 s = *(const float4*)&scores[(size_t)n * 4];
    float4 m = *(const float4*)&gmax[(size_t)g * 4];
    float4 d = *(const float4*)&gden[(size_t)g * 4];
    const float a0 = __expf(s.x - m.x) / d.x;
    const float a1 = __expf(s.y - m.y) / d.y;
    const float a2 = __expf(s.z - m.z) / d.z;
    const float a3 = __expf(s.w - m.w) / d.w;
    attn_out[(size_t)0 * N + n] = a0;
    attn_out[(size_t)1 * N + n] = a1;
    attn_out[(size_t)2 * N + n] = a2;
    attn_out[(size_t)3 * N + n] = a3;
    wsum[n] = invH * (a0 + a1 + a2 + a3);
}

// Kernel 4: pooled[g][d] = sum_{n in g} wsum[n] * x[n][d]   (one block per graph)
__global__ void pool_kernel(const float* __restrict__ x,
                            const int* __restrict__ bi,
                            const float* __restrict__ wsum,
                            float* __restrict__ pooled,
                            int N, int D) {
    const int g = blockIdx.x;
    const int d = threadIdx.x;                    // blockDim.x == D
    const int start = lower_bound_i(bi, N, g);
    const int end   = lower_bound_i(bi, N, g + 1);
    float acc = 0.f;
    int n = start;
    for (; n + 2 <= end; n += 2) {
        if (n + 8 < end)
            __builtin_prefetch(&x[(size_t)(n + 8) * D + d], 0, 1);
        const float w0 = wsum[n];
        const float w1 = wsum[n + 1];
        acc = fmaf(w0, x[(size_t)n * D + d], acc);
        acc = fmaf(w1, x[(size_t)(n + 1) * D + d], acc);
    }
    if (n < end)
        acc = fmaf(wsum[n], x[(size_t)n * D + d], acc);
    pooled[(size_t)g * D + d] = acc;              // zeros for empty graphs
}

// -----------------------------------------------------------------------------
extern "C" void kernel_launch(void* const* d_in, const int* in_sizes, int n_in,
                              void* d_out, int out_size, void* d_ws, size_t ws_size,
                              hipStream_t stream) {
    const float* x    = (const float*)d_in[0];
    const int*   bi   = (const int*)d_in[1];     // sorted batch indices
    // d_in[2] = num_graphs (device scalar; G derived on host below)
    const float* W    = (const float*)d_in[3];
    const float* bias = (const float*)d_in[4];
    const float* temp = (const float*)d_in[5];

    const int N = in_sizes[1];                   // 250000
    const int D = in_sizes[0] / N;               // 256
    const int H = in_sizes[4];                   // 4
    const int G = (int)((out_size - (long long)H * N) / D);   // 8192

    float* pooled   = (float*)d_out;                         // [G, D]
    float* attn_out = (float*)d_out + (size_t)G * D;         // [H, N]

    float* scores = (float*)d_ws;                            // [N, H]
    float* gmax   = scores + (size_t)N * H;                  // [G, H]
    float* gden   = gmax + (size_t)G * H;                    // [G, H]
    float* wsum   = gden + (size_t)G * H;                    // [N]

    const int chunks  = (N + 15) / 16;
    const int blocks1 = (chunks + K1_WAVES - 1) / K1_WAVES;
    scores_wmma_kernel<<<blocks1, K1_WAVES * 32, 0, stream>>>(x, W, bias, temp,
                                                              scores, N, D, H);
    seg_stats_kernel<<<G, 32, 0, stream>>>(scores, bi, gmax, gden, N);
    attn_kernel<<<(N + 255) / 256, 256, 0, stream>>>(scores, bi, gmax, gden,
                                                     attn_out, wsum, N, 1.0f / H);
    pool_kernel<<<G, D, 0, stream>>>(x, bi, wsum, pooled, N, D);
}